// GraphConv_45552423141527
// MI455X (gfx1250) — compile-verified
//
#include <hip/hip_runtime.h>
#include <stdint.h>

// GCNConv: out = D^-1/2 (A+I) D^-1/2 X W + b
// N=100000 nodes, C=128 channels, E=1.6M edges.
// Edge gather/scatter dominates (xw and out both fit in 192MB L2, so atomics
// resolve at L2). GEMM is 3.3 GFLOP -> use exact f32 WMMA (16x16x4_f32).

typedef __attribute__((ext_vector_type(2))) float v2f;
typedef __attribute__((ext_vector_type(8))) float v8f;

#define CH 128  // IN_C == OUT_C

// ---------- degree init: deg = 1 (self loop) ----------
__global__ void k_init_deg(float* __restrict__ deg, int n) {
    int i = blockIdx.x * blockDim.x + threadIdx.x;
    if (i < n) deg[i] = 1.0f;
}

// ---------- degree count at destination ----------
__global__ void k_deg_count(const int* __restrict__ dst, int E,
                            float* __restrict__ deg, int n) {
    int e = blockIdx.x * blockDim.x + threadIdx.x;
    if (e >= E) return;
    unsigned d = (unsigned)dst[e];
    if (d < (unsigned)n) unsafeAtomicAdd(&deg[d], 1.0f);  // global_atomic_add_f32
}

// ---------- dinv = rsqrt(deg), in place (deg >= 1 always) ----------
__global__ void k_dinv(float* __restrict__ deg, int n) {
    int i = blockIdx.x * blockDim.x + threadIdx.x;
    if (i < n) deg[i] = rsqrtf(deg[i]);
}

// ---------- xw = X @ W via V_WMMA_F32_16X16X4_F32 ----------
// One wave32 per 16x16 output tile. Block = 256 threads = 8 waves = the 8
// column tiles of one 16-row stripe. ISA VGPR layouts (CDNA5 ISA):
//   A 16x4 : lanes 0-15 -> K={0,1}, lanes 16-31 -> K={2,3}; M = lane%16
//   B 4x16 : mirrored (row K striped across lanes, N = lane%16)
//   C/D    : VGPR r -> M = r + 8*(lane>=16), N = lane%16
__global__ void k_gemm_wmma(const float* __restrict__ x, const float* __restrict__ W,
                            float* __restrict__ xw, int n) {
    const int warp = threadIdx.x >> 5;     // 0..7 -> column tile
    const int lane = threadIdx.x & 31;
    const int half = lane >> 4;            // 0 or 1
    const int lm   = lane & 15;
    const int row0 = blockIdx.x * 16;
    if (row0 >= n) return;                 // wave-uniform exit (EXEC stays full)
    const int col0 = warp * 16;
    const int kOff = half * 2;

    // Clamp row for a possible tail tile; keeps EXEC all-1s for WMMA.
    const int arow_i = (row0 + lm < n) ? (row0 + lm) : (n - 1);
    const float* __restrict__ arow = x + (size_t)arow_i * CH;

    v8f acc0 = {}; v8f acc1 = {};          // 2 accumulators: break RAW chain
    #pragma unroll
    for (int k = 0; k < CH; k += 8) {
        v2f a0 = *(const v2f*)(arow + k + kOff);          // K contiguous: 8B load
        v2f b0;
        b0.x = W[(size_t)(k + kOff)     * CH + col0 + lm];
        b0.y = W[(size_t)(k + kOff + 1) * CH + col0 + lm];
        acc0 = __builtin_amdgcn_wmma_f32_16x16x4_f32(
            false, a0, false, b0, (short)0, acc0, false, false);

        v2f a1 = *(const v2f*)(arow + k + 4 + kOff);
        v2f b1;
        b1.x = W[(size_t)(k + 4 + kOff)     * CH + col0 + lm];
        b1.y = W[(size_t)(k + 4 + kOff + 1) * CH + col0 + lm];
        acc1 = __builtin_amdgcn_wmma_f32_16x16x4_f32(
            false, a1, false, b1, (short)0, acc1, false, false);
    }
    acc0 += acc1;

    #pragma unroll
    for (int r = 0; r < 8; ++r) {
        int row = row0 + half * 8 + r;
        if (row < n) xw[(size_t)row * CH + col0 + lm] = acc0[r];
    }
}

// ---------- out = xw * dinv^2 (self loop) + b ----------
__global__ void k_init_out(const float* __restrict__ xw, const float* __restrict__ dinv,
                           const float* __restrict__ bias, float* __restrict__ out, int n) {
    int t = blockIdx.x * blockDim.x + threadIdx.x;     // one float4 each
    int total = n * (CH / 4);
    if (t >= total) return;
    int node = t >> 5;                                  // CH/4 == 32 groups/node
    int c4   = (t & 31) * 4;
    float s = dinv[node]; s = s * s;
    float4 v  = *(const float4*)(xw + (size_t)node * CH + c4);
    float4 bb = *(const float4*)(bias + c4);
    float4 o;
    o.x = v.x * s + bb.x;  o.y = v.y * s + bb.y;
    o.z = v.z * s + bb.z;  o.w = v.w * s + bb.w;
    *(float4*)(out + (size_t)node * CH + c4) = o;
}

// ---------- per-edge gather * norm, scatter-add at dst ----------
// One wave32 per edge; lane owns 4 contiguous channels -> coalesced b128 load
// of the full 512B row, then 4 f32 atomics that resolve in L2.
__global__ void k_scatter(const int* __restrict__ src, const int* __restrict__ dst, int E,
                          const float* __restrict__ xw, const float* __restrict__ dinv,
                          float* __restrict__ out, int n) {
    int gw   = (blockIdx.x * blockDim.x + threadIdx.x) >> 5;
    int lane = threadIdx.x & 31;
    if (gw >= E) return;
    unsigned s = (unsigned)src[gw];
    unsigned d = (unsigned)dst[gw];
    if (s >= (unsigned)n || d >= (unsigned)n) return;
    float norm = dinv[s] * dinv[d];
    float4 v = *(const float4*)(xw + (size_t)s * CH + lane * 4);
    float* o = out + (size_t)d * CH + lane * 4;
    unsafeAtomicAdd(o + 0, v.x * norm);
    unsafeAtomicAdd(o + 1, v.y * norm);
    unsafeAtomicAdd(o + 2, v.z * norm);
    unsafeAtomicAdd(o + 3, v.w * norm);
}

extern "C" void kernel_launch(void* const* d_in, const int* in_sizes, int n_in,
                              void* d_out, int out_size, void* d_ws, size_t ws_size,
                              hipStream_t stream) {
    // setup_inputs order: x [N*128] f32, edge_index [2*E] int, W [128*128] f32, b [128] f32
    const float* x    = (const float*)d_in[0];
    const int*   ei   = (const int*)  d_in[1];   // harness passes integers as int32
    const float* W    = (const float*)d_in[2];
    const float* bias = (const float*)d_in[3];
    float*       out  = (float*)d_out;

    const int n = in_sizes[0] / CH;
    const int E = in_sizes[1] / 2;
    const int* src = ei;
    const int* dst = ei + E;

    // workspace: [xw: n*CH f32][dinv: n f32]  (~51.6 MB)
    float* xw   = (float*)d_ws;
    float* dinv = xw + (size_t)n * CH;

    const int T = 256;
    k_init_deg <<<(n + T - 1) / T, T, 0, stream>>>(dinv, n);
    k_deg_count<<<(E + T - 1) / T, T, 0, stream>>>(dst, E, dinv, n);
    k_dinv     <<<(n + T - 1) / T, T, 0, stream>>>(dinv, n);

    k_gemm_wmma<<<(n + 15) / 16, 256, 0, stream>>>(x, W, xw, n);

    const int total4 = n * (CH / 4);
    k_init_out <<<(total4 + T - 1) / T, T, 0, stream>>>(xw, dinv, bias, out, n);

    const long long sthreads = (long long)E * 32;
    k_scatter  <<<(int)((sthreads + T - 1) / T), T, 0, stream>>>(src, dst, E, xw, dinv, out, n);
}